// SocialCircleLayer_89429809037696
// MI455X (gfx1250) — compile-verified
//
#include <hip/hip_runtime.h>
#include <math.h>

typedef __attribute__((ext_vector_type(2))) float v2f;
typedef __attribute__((ext_vector_type(8))) float v8f;

#define SC_MU      1e-4f
#define SC_EPS     1e-4f
#define SC_TWO_PI  6.28318530717958647692f

// One wave32 per batch element b.
//   Phase 1: stream nei_trajs[b] (N x 8 x 2 f32), compute per-neighbor
//            features + direction + angle bucket; stage in LDS.
//   Phase 2: 16x16x4 f32 WMMA accumulation of one-hot(bucket) x feats
//            (feature col 3 == 1.0 gives the per-partition counts).
__global__ __launch_bounds__(32) void social_circle_wmma_kernel(
    const float* __restrict__ trajs,   // [B, 8, 2]
    const float* __restrict__ nei,     // [B, N, 8, 2]
    const int*   __restrict__ part,    // scalar: partitions (<= 8 supported)
    float* __restrict__ out_sc,        // [B, P, 3]
    float* __restrict__ out_dir,       // [B, N]
    int N)
{
    const int b    = blockIdx.x;
    const int lane = threadIdx.x;              // 0..31
    const int P    = *part;

    __shared__ float sFeat[128][4];            // f_speed, f_dist, f_dir, 1.0
    __shared__ int   sIdx[128];                // angle bucket, -1 if invalid

    // Target trajectory: first & last positions.
    const float* tb  = trajs + (size_t)b * 16;
    const float t7x  = tb[14], t7y = tb[15];
    const float obx  = t7x - tb[0];
    const float oby  = t7y - tb[1];
    const float obs_den  = sqrtf(obx * obx + oby * oby) + SC_MU;
    const float idx_scale = (float)P * (1.0f / SC_TWO_PI);   // 1 / (2pi/P)

    const float* nb = nei + (size_t)b * (size_t)N * 16;

    const int per_lane = N >> 5;               // neighbors per lane (N=128 -> 4)
    for (int j = 0; j < per_lane; ++j) {
        const int n = j * 32 + lane;
        const float4* q = (const float4*)(nb + (size_t)n * 16);
        float4 q0 = q[0], q1 = q[1], q2 = q[2], q3 = q[3];

        // Validity mask: sum over all 16 samples != 0.
        float s = (q0.x + q0.y) + (q0.z + q0.w)
                + (q1.x + q1.y) + (q1.z + q1.w)
                + (q2.x + q2.y) + (q2.z + q2.w)
                + (q3.x + q3.y) + (q3.z + q3.w);

        // nei_vec = last - first
        float nvx = q3.z - q0.x;
        float nvy = q3.w - q0.y;
        float f_speed = (sqrtf(nvx * nvx + nvy * nvy) + SC_MU) / obs_den;

        // nei_pos = nei last - target last
        float px = q3.z - t7x;
        float py = q3.w - t7y;
        float f_dist = sqrtf(px * px + py * py);

        float d = atan2f(py, px);
        d -= SC_TWO_PI * floorf(d * (1.0f / SC_TWO_PI));   // mod 2pi, in [0,2pi)

        int idx = (int)(d * idx_scale);        // truncation, d >= 0
        if (s == 0.0f) idx = -1;               // invalid neighbor -> no bucket

        out_dir[(size_t)b * N + n] = d;

        sFeat[n][0] = f_speed;
        sFeat[n][1] = f_dist;
        sFeat[n][2] = d;
        sFeat[n][3] = 1.0f;                    // count column
        sIdx[n]     = idx;
    }
    __syncthreads();

    // D[p][f] = sum_n (idx[n]==p) * feat[n][f]  via V_WMMA_F32_16X16X4_F32.
    // A 16x4 f32 layout: lanes 0-15 rows M=0..15 with K=0(v0),1(v1);
    //                    lanes 16-31 same rows with K=2(v0),3(v1).
    // B 4x16 f32 layout mirrored: lane&15 = column, lane>>4 selects K 0/1 vs 2/3.
    const int m    = lane & 15;                // A row (partition) / B column (feature)
    const int f    = m & 3;                    // in-bounds feature index for LDS read
    const int koff = (lane >> 4) * 2;

    v8f acc = {0.f, 0.f, 0.f, 0.f, 0.f, 0.f, 0.f, 0.f};
    for (int base = 0; base < N; base += 4) {
        int i0 = sIdx[base + koff + 0];
        int i1 = sIdx[base + koff + 1];
        v2f a;
        a.x = (i0 == m) ? 1.0f : 0.0f;
        a.y = (i1 == m) ? 1.0f : 0.0f;

        float b0 = sFeat[base + koff + 0][f];
        float b1 = sFeat[base + koff + 1][f];
        v2f bm;
        bm.x = (m < 4) ? b0 : 0.0f;
        bm.y = (m < 4) ? b1 : 0.0f;

        acc = __builtin_amdgcn_wmma_f32_16x16x4_f32(
            false, a, false, bm, (short)0, acc, false, false);
    }

    // Row p of D sits in acc[p], lanes 0-15 = columns 0-15.
    // Columns 0..2 = feature sums, column 3 = count.
    #pragma unroll
    for (int p = 0; p < 8; ++p) {
        if (p >= P) break;
        float v   = acc[p];
        float cnt = __shfl(v, 3, 32) + SC_EPS;
        if (lane < 3)
            out_sc[((size_t)b * P + p) * 3 + lane] = v / cnt;
    }
}

extern "C" void kernel_launch(void* const* d_in, const int* in_sizes, int n_in,
                              void* d_out, int out_size, void* d_ws, size_t ws_size,
                              hipStream_t stream) {
    const float* trajs = (const float*)d_in[0];
    const float* nei   = (const float*)d_in[1];
    const int*   part  = (const int*)d_in[2];

    const int B = in_sizes[0] / 16;            // trajs [B, 8, 2]
    const int N = in_sizes[1] / (B * 16);      // nei   [B, N, 8, 2]
    const int P = (out_size - B * N) / (3 * B);// out = B*P*3 + B*N

    float* out_sc  = (float*)d_out;
    float* out_dir = out_sc + (size_t)B * P * 3;

    social_circle_wmma_kernel<<<B, 32, 0, stream>>>(trajs, nei, part,
                                                    out_sc, out_dir, N);
}